// MultiSimilarityLoss_52381421142559
// MI455X (gfx1250) — compile-verified
//
#include <hip/hip_runtime.h>
#include <hip/hip_bf16.h>
#include <math.h>

#define B_ROWS   8192
#define BLK      256
#define WAVES    (BLK / 32)
#define NV       (B_ROWS / (BLK * 4))   // 8 vec4 chunks per thread
#define MARGIN   0.1f
#define THRESH   0.5f
#define SCALE_POS 2.0f
#define SCALE_NEG 40.0f
#define EPS      1e-5f

#if defined(__gfx1250__) && __has_builtin(__builtin_amdgcn_global_load_async_to_lds_b128)
#define HAVE_ASYNC_LDS 1
#else
#define HAVE_ASYNC_LDS 0
#endif

typedef __attribute__((ext_vector_type(4))) float v4f;
typedef __attribute__((ext_vector_type(4))) int   v4i;
// Pointer-to-AS1 / pointer-to-AS3 v4i, matching the async-LDS builtin params.
typedef __attribute__((address_space(1))) v4i* gv4i_p;
typedef __attribute__((address_space(3))) v4i* lv4i_p;

__device__ __forceinline__ float wave_min(float v) {
#pragma unroll
  for (int o = 16; o > 0; o >>= 1) v = fminf(v, __shfl_xor(v, o, 32));
  return v;
}
__device__ __forceinline__ float wave_max(float v) {
#pragma unroll
  for (int o = 16; o > 0; o >>= 1) v = fmaxf(v, __shfl_xor(v, o, 32));
  return v;
}
__device__ __forceinline__ float wave_add(float v) {
#pragma unroll
  for (int o = 16; o > 0; o >>= 1) v += __shfl_xor(v, o, 32);
  return v;
}

__global__ __launch_bounds__(BLK) void ms_row_kernel(
    const float* __restrict__ sim, const int* __restrict__ labels,
    float* __restrict__ row_loss) {
  __shared__ __align__(16) float s_row[B_ROWS];   // 32 KB: one sim row
  __shared__ unsigned s_mask4[B_ROWS / 4];        // 8 KB: byte-packed same-mask
  __shared__ float s_redA[WAVES];
  __shared__ float s_redB[WAVES];
  __shared__ float s_bc[2];

  const int row  = blockIdx.x;
  const int tid  = threadIdx.x;
  const int lane = tid & 31;
  const int wid  = tid >> 5;
  const float* rowp = sim + (size_t)row * B_ROWS;

  // ---- Kick off async DMA of the row into LDS (8 x b128 per thread) ----
#pragma unroll
  for (int k = 0; k < NV; ++k) {
    const int vi = tid + k * BLK;                 // vec4 index
#if HAVE_ASYNC_LDS
    __builtin_amdgcn_global_load_async_to_lds_b128(
        (gv4i_p)((const v4f*)rowp + vi),
        (lv4i_p)((v4f*)s_row + vi),
        0, 0);
#else
    ((v4f*)s_row)[vi] = ((const v4f*)rowp)[vi];
#endif
  }

  // ---- Overlap with the DMA: build byte-packed same-label mask in LDS.
  // labels (32 KB total) is L2-resident across all 8192 blocks, so this is
  // pure L2 traffic hidden under the HBM stream.
  const int myLab = labels[row];
#pragma unroll
  for (int k = 0; k < NV; ++k) {
    const int vi = tid + k * BLK;
    const v4i lab4 = ((const v4i*)labels)[vi];
    unsigned m = 0u;
    if (lab4.x == myLab) m |= 0x00000001u;
    if (lab4.y == myLab) m |= 0x00000100u;
    if (lab4.z == myLab) m |= 0x00010000u;
    if (lab4.w == myLab) m |= 0x01000000u;
    s_mask4[vi] = m;
  }

#if HAVE_ASYNC_LDS
#if __has_builtin(__builtin_amdgcn_s_wait_asynccnt)
  __builtin_amdgcn_s_wait_asynccnt(0);
#else
  asm volatile("s_wait_asynccnt 0" ::: "memory");
#endif
#endif
  __syncthreads();

  // ---- Pass 1: hardest positive (min) / hardest negative (max), vec4-wide ----
  float minPos = INFINITY;
  float maxNeg = -INFINITY;
#pragma unroll
  for (int k = 0; k < NV; ++k) {
    const int vi = tid + k * BLK;
    const v4f s = ((const v4f*)s_row)[vi];        // ds_load_b128
    const unsigned m = s_mask4[vi];
#pragma unroll
    for (int c = 0; c < 4; ++c) {
      const float sc = s[c];
      if ((m >> (8 * c)) & 1u) {
        if (sc < 1.0f - EPS) minPos = fminf(minPos, sc);
      } else {
        maxNeg = fmaxf(maxNeg, sc);
      }
    }
  }
  const float wmin = wave_min(minPos);
  const float wmax = wave_max(maxNeg);
  if (lane == 0) { s_redA[wid] = wmin; s_redB[wid] = wmax; }
  __syncthreads();
  if (tid == 0) {
    float a = s_redA[0], b = s_redB[0];
#pragma unroll
    for (int i = 1; i < WAVES; ++i) { a = fminf(a, s_redA[i]); b = fmaxf(b, s_redB[i]); }
    s_bc[0] = a; s_bc[1] = b;
  }
  __syncthreads();
  const float minPosRow = s_bc[0];
  const float maxNegRow = s_bc[1];

  // ---- Pass 2: mined exp-sums (LDS-only reads), vec4-wide ----
  float posSum = 0.0f, negSum = 0.0f;
#pragma unroll
  for (int k = 0; k < NV; ++k) {
    const int vi = tid + k * BLK;
    const v4f s = ((const v4f*)s_row)[vi];
    const unsigned m = s_mask4[vi];
#pragma unroll
    for (int c = 0; c < 4; ++c) {
      const float sc = s[c];
      const bool same = (m >> (8 * c)) & 1u;
      const bool selPos = same && (sc < 1.0f - EPS) && (sc - MARGIN < maxNegRow);
      const bool selNeg = (!same) && (sc + MARGIN > minPosRow);
      if (selPos) posSum += __expf(-SCALE_POS * (sc - THRESH));
      if (selNeg) negSum += __expf(SCALE_NEG * (sc - THRESH));
    }
  }
  posSum = wave_add(posSum);
  negSum = wave_add(negSum);
  if (lane == 0) { s_redA[wid] = posSum; s_redB[wid] = negSum; }
  __syncthreads();
  if (tid == 0) {
    float ps = 0.0f, ns = 0.0f;
#pragma unroll
    for (int i = 0; i < WAVES; ++i) { ps += s_redA[i]; ns += s_redB[i]; }
    // exp terms are strictly positive (smallest selectable term ~e^-60, far
    // above f32 min normal), so sum > 0 <=> any selected pair: matches `valid`.
    const bool valid = (ps > 0.0f) && (ns > 0.0f);
    row_loss[row] = valid
        ? (log1pf(ps) * (1.0f / SCALE_POS) + log1pf(ns) * (1.0f / SCALE_NEG))
        : 0.0f;
  }
}

__global__ __launch_bounds__(BLK) void ms_final_kernel(
    const float* __restrict__ row_loss, float* __restrict__ out) {
  __shared__ float s_red[WAVES];
  float s = 0.0f;
  for (int j = threadIdx.x; j < B_ROWS; j += BLK) s += row_loss[j];
  s = wave_add(s);
  const int lane = threadIdx.x & 31, wid = threadIdx.x >> 5;
  if (lane == 0) s_red[wid] = s;
  __syncthreads();
  if (threadIdx.x == 0) {
    float t = 0.0f;
#pragma unroll
    for (int i = 0; i < WAVES; ++i) t += s_red[i];
    out[0] = t * (1.0f / (float)B_ROWS);
  }
}

extern "C" void kernel_launch(void* const* d_in, const int* in_sizes, int n_in,
                              void* d_out, int out_size, void* d_ws, size_t ws_size,
                              hipStream_t stream) {
  (void)in_sizes; (void)n_in; (void)out_size; (void)ws_size;
  const float* sim    = (const float*)d_in[0];
  const int*   labels = (const int*)d_in[1];
  float*       ws     = (float*)d_ws;       // 8192 per-row losses (32 KB)
  ms_row_kernel<<<B_ROWS, BLK, 0, stream>>>(sim, labels, ws);
  ms_final_kernel<<<1, BLK, 0, stream>>>(ws, (float*)d_out);
}